// DevNet_61967788147232
// MI455X (gfx1250) — compile-verified
//
#include <hip/hip_runtime.h>
#include <hip/hip_bf16.h>
#include <cstdint>

#define B_    8
#define H_    512
#define W_    512
#define HO_   510
#define WO_   510
#define NPIX  (HO_ * WO_)      // 260100, divisible by 4
#define NOFFC 18
#define NMSKC 9
#define NCH   (NOFFC + NMSKC)  // 27 staged channels
#define TPB   256
#define NCHUNK (NCH * (TPB / 4))   // 1728 x 16B chunks per block tile

typedef __attribute__((address_space(3))) float lds_float;

__device__ __forceinline__ float corner_fetch(const float* __restrict__ img,
                                              float yc, float xc) {
  // reference: valid = (yc in [0,H-1]) & (xc in [0,W-1]); gather at clipped idx; zero if invalid
  bool ok = (yc >= 0.0f) && (yc <= (float)(H_ - 1)) &&
            (xc >= 0.0f) && (xc <= (float)(W_ - 1));
  int yi = (int)fminf(fmaxf(yc, 0.0f), (float)(H_ - 1));
  int xi = (int)fminf(fmaxf(xc, 0.0f), (float)(W_ - 1));
  float v = img[(yi << 9) + xi];       // W_ == 512 -> shift; default RT policy, stays hot in L2/WGP$
  return ok ? v : 0.0f;
}

__device__ __forceinline__ void async_stage_nt(unsigned lds_addr, const float* src) {
  // One-pass stream: non-temporal so 225 MB of offset/mask doesn't evict the hot image from L2.
  asm volatile("global_load_async_to_lds_b128 %0, %1, off th:TH_LOAD_NT"
               :: "v"(lds_addr), "v"((unsigned long long)(uintptr_t)src) : "memory");
}

__global__ __launch_bounds__(TPB)
void deform_conv2d_kernel(const float* __restrict__ x,
                          const float* __restrict__ offset,
                          const float* __restrict__ mask,
                          const float* __restrict__ weight,
                          const float* __restrict__ bias,
                          float* __restrict__ out) {
  __shared__ __align__(16) float tile[NCH * TPB];   // 27 KB: offset[18]+mask[9] per pixel

  const int b   = blockIdx.y;
  const int p0  = blockIdx.x * TPB;                 // first pixel of this block's tile
  const int tid = threadIdx.x;

  // Issue uniform weight/bias scalar loads early; they overlap the async staging.
  float w0[9], w1[9], w2[9];
#pragma unroll
  for (int k = 0; k < 9; ++k) {
    w0[k] = weight[k];
    w1[k] = weight[9 + k];
    w2[k] = weight[18 + k];
  }
  const float bias0 = bias[0], bias1 = bias[1], bias2 = bias[2];

  // ---- Stage the streaming offset/mask tile into LDS with CDNA5 async copies.
  // 27 channels * 256 pixels * 4B = 1728 16B chunks; lanes 0..191 move 7, rest 6.
  const bool full = (p0 + TPB) <= NPIX;             // uniform: true for 8128/8136 blocks
#pragma unroll
  for (int j = 0; j < 7; ++j) {
    const int i = tid + j * TPB;
    if (i < NCHUNK) {                               // folds away for j < 6
      const int c   = i >> 6;                       // channel 0..26
      const int q   = (i & 63) << 2;                // pixel offset within tile (mult of 4)
      const int pix = p0 + q;
      if (full || pix < NPIX) {                     // planes tile exactly by 16B chunks
        const float* src = (c < NOFFC)
            ? offset + ((size_t)b * NOFFC + c) * NPIX + pix
            : mask   + ((size_t)b * NMSKC + (c - NOFFC)) * NPIX + pix;
        unsigned lds_addr = (unsigned)(uintptr_t)(lds_float*)&tile[c * TPB + q];
        async_stage_nt(lds_addr, src);
      }
    }
  }
  asm volatile("s_wait_asynccnt 0x0" ::: "memory"); // this wave's async copies complete
  __syncthreads();                                   // tile visible to whole workgroup

  const int p = p0 + tid;
  if (p >= NPIX) return;

  const float* xb = x + (size_t)b * (H_ * W_);
  const int ho = p / WO_;
  const int wo = p - ho * WO_;

  float acc0 = 0.0f, acc1 = 0.0f, acc2 = 0.0f;
#pragma unroll
  for (int k = 0; k < 9; ++k) {
    const float offy = tile[(2 * k)     * TPB + tid];   // dy = channel 2k
    const float offx = tile[(2 * k + 1) * TPB + tid];   // dx = channel 2k+1
    const float m    = tile[(NOFFC + k) * TPB + tid];

    const float py = offy + (float)ho + (float)(k / 3);
    const float px = offx + (float)wo + (float)(k % 3);

    const float y0 = floorf(py);
    const float x0 = floorf(px);
    const float ly = py - y0;
    const float lx = px - x0;

    const float v00 = corner_fetch(xb, y0,        x0);
    const float v01 = corner_fetch(xb, y0,        x0 + 1.0f);
    const float v10 = corner_fetch(xb, y0 + 1.0f, x0);
    const float v11 = corner_fetch(xb, y0 + 1.0f, x0 + 1.0f);

    float s = v00 * (1.0f - ly) * (1.0f - lx)
            + v01 * (1.0f - ly) * lx
            + v10 * ly * (1.0f - lx)
            + v11 * ly * lx;
    s *= m;

    acc0 = fmaf(w0[k], s, acc0);
    acc1 = fmaf(w1[k], s, acc1);
    acc2 = fmaf(w2[k], s, acc2);
  }

  // Output is write-once: non-temporal stores keep the 25 MB out-stream from
  // occupying L2 lines the x-gathers want.
  const size_t ob = (size_t)b * 3 * NPIX + p;          // out [B,3,HO,WO]
  __builtin_nontemporal_store(acc0 + bias0, &out[ob]);
  __builtin_nontemporal_store(acc1 + bias1, &out[ob + NPIX]);
  __builtin_nontemporal_store(acc2 + bias2, &out[ob + 2 * NPIX]);
}

extern "C" void kernel_launch(void* const* d_in, const int* in_sizes, int n_in,
                              void* d_out, int out_size, void* d_ws, size_t ws_size,
                              hipStream_t stream) {
  const float* x      = (const float*)d_in[0];
  const float* offset = (const float*)d_in[1];
  const float* mask   = (const float*)d_in[2];
  const float* weight = (const float*)d_in[3];
  const float* bias   = (const float*)d_in[4];
  float* out = (float*)d_out;

  dim3 grid((NPIX + TPB - 1) / TPB, B_);   // 1017 x 8 blocks
  deform_conv2d_kernel<<<grid, TPB, 0, stream>>>(x, offset, mask, weight, bias, out);
}